// GraphEncoder_18528488915234
// MI455X (gfx1250) — compile-verified
//
#include <hip/hip_runtime.h>
#include <math.h>

#define N_NODES 20000
#define N_EDGES 640000
#define IN_DIM  128
#define HEADS   4
#define CDIM    64
#define OUTD    256     // HEADS*CDIM
#define EDIM    16
#define NLAYERS 3
#define NGRAPH  64
#define NEG_SLOPE 0.2f
#define LN_EPS  1e-5f
#define ENC_NEG_INF 0x007FFFFFu   // encf(-inf)

typedef float v2f __attribute__((ext_vector_type(2)));
typedef float v8f __attribute__((ext_vector_type(8)));

// ---- order-preserving float <-> uint encoding for atomic max ----
__device__ __forceinline__ unsigned encf(float x) {
  unsigned b = __float_as_uint(x);
  return (b & 0x80000000u) ? ~b : (b | 0x80000000u);
}
__device__ __forceinline__ float decf(unsigned e) {
  unsigned b = (e & 0x80000000u) ? (e & 0x7FFFFFFFu) : ~e;
  return __uint_as_float(b);
}

// ---------------- h = X @ W via V_WMMA_F32_16X16X4_F32 ----------------
// grid = (N/16, 256/16), block = 32 (one wave). X:[N,K], W:[K,256], H:[N,256]
__global__ void gemm_wmma(const float* __restrict__ X, const float* __restrict__ W,
                          float* __restrict__ Hout, int K) {
  const int tm = blockIdx.x;
  const int tn = blockIdx.y;
  const int lane = threadIdx.x;
  const int hl  = lane >> 4;        // 0: K pair {0,1}, 1: K pair {2,3}
  const int l16 = lane & 15;
  const int row = tm * 16 + l16;    // A: lane -> M row
  const int col = tn * 16 + l16;    // B: lane -> N col
  v8f acc = {};
  const float* xr = X + (size_t)row * K + hl * 2;
  const float* wc = W + (size_t)(hl * 2) * OUTD + col;
  for (int k = 0; k < K; k += 4) {
    v2f a, b;
    a.x = xr[k];
    a.y = xr[k + 1];
    b.x = wc[(size_t)k * OUTD];
    b.y = wc[(size_t)(k + 1) * OUTD];
    acc = __builtin_amdgcn_wmma_f32_16x16x4_f32(false, a, false, b,
                                                (short)0, acc, false, false);
  }
  // D layout: VGPR r -> (M = r + 8*hl, N = l16)
  float* out = Hout + (size_t)(tm * 16 + hl * 8) * OUTD + tn * 16 + l16;
#pragma unroll
  for (int r = 0; r < 8; ++r) out[(size_t)r * OUTD] = acc[r];
}

// ---------------- M[16][4] = reduce(W_e, att_edge) ----------------
__global__ void edge_att_mat(const float* __restrict__ We,
                             const float* __restrict__ att_edge,
                             float* __restrict__ Mmat) {
  int t = threadIdx.x;              // 64 threads
  int ed = t >> 2, hh = t & 3;
  float s = 0.f;
  for (int c = 0; c < CDIM; ++c)
    s += We[ed * OUTD + hh * CDIM + c] * att_edge[hh * CDIM + c];
  Mmat[ed * 4 + hh] = s;
}

// -------- per (node,head): att dots + init of max/sum/accumulator --------
__global__ void node_att_init(const float* __restrict__ H,
                              const float* __restrict__ att_src,
                              const float* __restrict__ att_dst,
                              float* __restrict__ a_src, float* __restrict__ a_dst,
                              unsigned* __restrict__ m_enc, float* __restrict__ ssum,
                              float* __restrict__ acc) {
  int t = blockIdx.x * blockDim.x + threadIdx.x;
  if (t >= N_NODES * HEADS) return;
  int n = t >> 2, hh = t & 3;
  const float4* hp = (const float4*)(H + (size_t)n * OUTD + hh * CDIM);
  const float4* as = (const float4*)(att_src + hh * CDIM);
  const float4* ad = (const float4*)(att_dst + hh * CDIM);
  float4* ap = (float4*)(acc + (size_t)n * OUTD + hh * CDIM);
  float s1 = 0.f, s2 = 0.f;
#pragma unroll
  for (int i = 0; i < 16; ++i) {
    float4 hv = hp[i], a1 = as[i], a2 = ad[i];
    s1 += hv.x * a1.x + hv.y * a1.y + hv.z * a1.z + hv.w * a1.w;
    s2 += hv.x * a2.x + hv.y * a2.y + hv.z * a2.z + hv.w * a2.w;
    ap[i] = make_float4(0.f, 0.f, 0.f, 0.f);
  }
  a_src[t] = s1;
  a_dst[t] = s2;
  m_enc[t] = ENC_NEG_INF;
  ssum[t] = 0.f;
}

// -------- per edge: alpha = lrelu(...), atomic segment max --------
__global__ void edge_alpha(const int* __restrict__ src, const int* __restrict__ dst,
                           const float* __restrict__ ea, const float* __restrict__ Mmat,
                           const float* __restrict__ a_src, const float* __restrict__ a_dst,
                           float* __restrict__ alpha, unsigned* __restrict__ m_enc) {
  int e = blockIdx.x * blockDim.x + threadIdx.x;
  if (e >= N_EDGES) return;
  int s0 = src[e], d0 = dst[e];
  float eav[EDIM];
  const float4* ep = (const float4*)(ea + (size_t)e * EDIM);
#pragma unroll
  for (int i = 0; i < 4; ++i) {
    float4 v = ep[i];
    eav[4 * i] = v.x; eav[4 * i + 1] = v.y; eav[4 * i + 2] = v.z; eav[4 * i + 3] = v.w;
  }
#pragma unroll
  for (int hh = 0; hh < HEADS; ++hh) {
    float ae = 0.f;
#pragma unroll
    for (int j = 0; j < EDIM; ++j) ae += eav[j] * Mmat[j * 4 + hh];
    float al = a_src[s0 * 4 + hh] + a_dst[d0 * 4 + hh] + ae;
    al = al > 0.f ? al : NEG_SLOPE * al;
    alpha[(size_t)e * 4 + hh] = al;
    atomicMax(&m_enc[d0 * 4 + hh], encf(al));
  }
}

// -------- per edge: exp(alpha - max), atomic segment sum --------
__global__ void edge_exp(const int* __restrict__ dst, const unsigned* __restrict__ m_enc,
                         float* __restrict__ alpha, float* __restrict__ ssum) {
  int e = blockIdx.x * blockDim.x + threadIdx.x;
  if (e >= N_EDGES) return;
  int d0 = dst[e];
#pragma unroll
  for (int hh = 0; hh < HEADS; ++hh) {
    float mx = decf(m_enc[d0 * 4 + hh]);
    float ex = __expf(alpha[(size_t)e * 4 + hh] - mx);
    alpha[(size_t)e * 4 + hh] = ex;
    atomicAdd(&ssum[d0 * 4 + hh], ex);
  }
}

// -------- one wave per edge: acc[dst] += h[src] * (ex / s[dst]) --------
__global__ void scatter_msg(const int* __restrict__ src, const int* __restrict__ dst,
                            const float* __restrict__ H, const float* __restrict__ alpha,
                            const float* __restrict__ ssum, float* __restrict__ acc) {
  int wid = blockIdx.x * 8 + (threadIdx.x >> 5);   // edge id, E = 80000*8
  int lane = threadIdx.x & 31;
  int s0 = src[wid], d0 = dst[wid];
  int head = lane >> 3;                            // 8 channels per lane, one head
  float w = alpha[(size_t)wid * 4 + head] / (ssum[d0 * 4 + head] + 1e-16f);
  const float4* hp = (const float4*)(H + (size_t)s0 * OUTD + lane * 8);
  float4 h0 = hp[0], h1 = hp[1];
  float* op = acc + (size_t)d0 * OUTD + lane * 8;
  atomicAdd(op + 0, h0.x * w); atomicAdd(op + 1, h0.y * w);
  atomicAdd(op + 2, h0.z * w); atomicAdd(op + 3, h0.w * w);
  atomicAdd(op + 4, h1.x * w); atomicAdd(op + 5, h1.y * w);
  atomicAdd(op + 6, h1.z * w); atomicAdd(op + 7, h1.w * w);
}

// -------- one wave per node: +bias, LayerNorm, ReLU --------
__global__ void ln_relu(const float* __restrict__ acc, const float* __restrict__ bias,
                        const float* __restrict__ gamma, const float* __restrict__ beta,
                        float* __restrict__ xout) {
  int node = blockIdx.x * 8 + (threadIdx.x >> 5);
  int lane = threadIdx.x & 31;
  int base = lane * 8;
  float v[8];
  float sum = 0.f, sq = 0.f;
#pragma unroll
  for (int i = 0; i < 8; ++i) {
    float x = acc[(size_t)node * OUTD + base + i] + bias[base + i];
    v[i] = x; sum += x; sq += x * x;
  }
#pragma unroll
  for (int m = 16; m >= 1; m >>= 1) {
    sum += __shfl_xor(sum, m, 32);
    sq  += __shfl_xor(sq,  m, 32);
  }
  float mu = sum * (1.f / OUTD);
  float var = sq * (1.f / OUTD) - mu * mu;
  float inv = rsqrtf(var + LN_EPS);
#pragma unroll
  for (int i = 0; i < 8; ++i) {
    float o = (v[i] - mu) * inv * gamma[base + i] + beta[base + i];
    xout[(size_t)node * OUTD + base + i] = fmaxf(o, 0.f);
  }
}

// ---------------- graph mean pooling ----------------
__global__ void pool_zero(float* __restrict__ gsum, float* __restrict__ gcnt) {
  int t = blockIdx.x * blockDim.x + threadIdx.x;
  if (t < NGRAPH * OUTD) gsum[t] = 0.f;
  if (t < NGRAPH) gcnt[t] = 0.f;
}
__global__ void pool_add(const float* __restrict__ x, const int* __restrict__ batch,
                         float* __restrict__ gsum, float* __restrict__ gcnt) {
  int t = blockIdx.x * blockDim.x + threadIdx.x;   // N_NODES * 64
  if (t >= N_NODES * 64) return;
  int n = t >> 6, c4 = t & 63;
  int b = batch[n];
  float4 v = *(const float4*)(x + (size_t)n * OUTD + c4 * 4);
  float* gp = gsum + (size_t)b * OUTD + c4 * 4;
  atomicAdd(gp + 0, v.x); atomicAdd(gp + 1, v.y);
  atomicAdd(gp + 2, v.z); atomicAdd(gp + 3, v.w);
  if (c4 == 0) atomicAdd(&gcnt[b], 1.f);
}
__global__ void pool_div(const float* __restrict__ gsum, const float* __restrict__ gcnt,
                         float* __restrict__ out) {
  int t = blockIdx.x * blockDim.x + threadIdx.x;
  if (t >= NGRAPH * OUTD) return;
  out[t] = gsum[t] / fmaxf(gcnt[t >> 8], 1.f);
}

extern "C" void kernel_launch(void* const* d_in, const int* in_sizes, int n_in,
                              void* d_out, int out_size, void* d_ws, size_t ws_size,
                              hipStream_t stream) {
  (void)in_sizes; (void)n_in; (void)out_size; (void)ws_size;
  const float* x_in  = (const float*)d_in[0];
  const int*   ei    = (const int*)d_in[1];
  const float* eattr = (const float*)d_in[2];
  const int*   batch = (const int*)d_in[3];
  const int* src = ei;
  const int* dst = ei + N_EDGES;

  char* ws = (char*)d_ws;
  size_t off = 0;
  auto carve = [&](size_t bytes) -> void* {
    void* p = ws + off;
    off += (bytes + 255) & ~(size_t)255;
    return p;
  };
  float*    h     = (float*)carve(sizeof(float) * (size_t)N_NODES * OUTD);
  float*    acc   = (float*)carve(sizeof(float) * (size_t)N_NODES * OUTD);
  float*    xA    = (float*)carve(sizeof(float) * (size_t)N_NODES * OUTD);
  float*    a_src = (float*)carve(sizeof(float) * N_NODES * HEADS);
  float*    a_dst = (float*)carve(sizeof(float) * N_NODES * HEADS);
  unsigned* m_enc = (unsigned*)carve(sizeof(unsigned) * N_NODES * HEADS);
  float*    ssum  = (float*)carve(sizeof(float) * N_NODES * HEADS);
  float*    alpha = (float*)carve(sizeof(float) * (size_t)N_EDGES * HEADS);
  float*    Mmat  = (float*)carve(sizeof(float) * EDIM * HEADS);
  float*    gsum  = (float*)carve(sizeof(float) * NGRAPH * OUTD);
  float*    gcnt  = (float*)carve(sizeof(float) * NGRAPH);

  float* outp = (float*)d_out;
  float* gemb = outp;                       // [G, 256]
  float* xfin = outp + NGRAPH * OUTD;       // [N, 256]

  const float* cur = x_in;
  int K = IN_DIM;
  for (int l = 0; l < NLAYERS; ++l) {
    const float* W    = (const float*)d_in[4 + l * 8 + 0];
    const float* Asrc = (const float*)d_in[4 + l * 8 + 1];
    const float* Adst = (const float*)d_in[4 + l * 8 + 2];
    const float* We   = (const float*)d_in[4 + l * 8 + 3];
    const float* Aedg = (const float*)d_in[4 + l * 8 + 4];
    const float* bias = (const float*)d_in[4 + l * 8 + 5];
    const float* gam  = (const float*)d_in[4 + l * 8 + 6];
    const float* bet  = (const float*)d_in[4 + l * 8 + 7];
    float* xout = (l == NLAYERS - 1) ? xfin : xA;

    dim3 gg(N_NODES / 16, OUTD / 16);
    gemm_wmma<<<gg, 32, 0, stream>>>(cur, W, h, K);
    edge_att_mat<<<1, 64, 0, stream>>>(We, Aedg, Mmat);
    node_att_init<<<(N_NODES * HEADS + 255) / 256, 256, 0, stream>>>(
        h, Asrc, Adst, a_src, a_dst, m_enc, ssum, acc);
    edge_alpha<<<(N_EDGES + 255) / 256, 256, 0, stream>>>(
        src, dst, eattr, Mmat, a_src, a_dst, alpha, m_enc);
    edge_exp<<<(N_EDGES + 255) / 256, 256, 0, stream>>>(dst, m_enc, alpha, ssum);
    scatter_msg<<<N_EDGES / 8, 256, 0, stream>>>(src, dst, h, alpha, ssum, acc);
    ln_relu<<<N_NODES / 8, 256, 0, stream>>>(acc, bias, gam, bet, xout);
    cur = xout;
    K = OUTD;
  }
  pool_zero<<<(NGRAPH * OUTD + 255) / 256, 256, 0, stream>>>(gsum, gcnt);
  pool_add<<<(N_NODES * 64 + 255) / 256, 256, 0, stream>>>(xfin, batch, gsum, gcnt);
  pool_div<<<(NGRAPH * OUTD + 255) / 256, 256, 0, stream>>>(gsum, gcnt, gemb);
}